// MVDR_oracle_using_steering_vector_68135361183976
// MI455X (gfx1250) — compile-verified
//
#include <hip/hip_runtime.h>
#include <hip/hip_bf16.h>

typedef __attribute__((ext_vector_type(2))) float v2f;
typedef __attribute__((ext_vector_type(4))) float v4f;
typedef __attribute__((ext_vector_type(8))) float v8f;

#define NN 8
#define CC 8
#define TT 600
#define FF 257
#define FTILE 32
#define TTILE 16
#define NFT 9                 // ceil(257/32)
#define FPAD 33
#define MVDR_LOAD 7.0710678118654752e-4f   // 0.001/sqrt(2)

// ---------------------------------------------------------------------------
// Kernel 1: per (n,f) 16x16 fp32 Gram of [Vr;Vi] (16 x 600) via
// v_wmma_f32_16x16x4_f32. One workgroup = (n, 32 consecutive f); 8 waves,
// each wave owns f_local in {w, w+8, w+16, w+24}.
// LDS tile layout: [p = tt/2][m][f][e = tt&1]  (t-pairs innermost) so each
// lane's (A[m][kb], A[m][kb+1]) is one aligned ds_load_b64 straight into the
// WMMA A/B source register pair. Row stride 33 keeps reads bank-conflict-free.
// ---------------------------------------------------------------------------
__global__ __launch_bounds__(256) void mvdr_cov(const float* __restrict__ v,
                                                float* __restrict__ G) {
  __shared__ float tile[8 * 16 * FPAD * 2];   // 8448 floats, 33 KB
  const int n    = blockIdx.x / NFT;
  const int f0   = (blockIdx.x % NFT) * FTILE;
  const int tid  = threadIdx.x;
  const int wave = tid >> 5;
  const int lane = tid & 31;
  const int mlane = lane & 15;
  const int kphalf = lane >> 4;          // 0 or 1 (K-pair offset for this half)

  v8f acc[4];
#pragma unroll
  for (int i = 0; i < 4; ++i)
#pragma unroll
    for (int j = 0; j < 8; ++j) acc[i][j] = 0.0f;

  for (int t0 = 0; t0 < TT; t0 += TTILE) {
    // cooperative coalesced load: lane -> f, rows (tt,m) strided across waves
    for (int r = wave; r < 256; r += 8) {
      const int m  = r & 15;             // m = ri*8 + c  (rows 0-7 Re, 8-15 Im)
      const int tt = r >> 4;
      const int t  = t0 + tt;
      const int f  = f0 + lane;
      float val = 0.0f;
      if (t < TT && f < FF)
        val = v[((size_t)(n * 16 + m) * TT + t) * FF + f];
      tile[((((tt >> 1) * 16 + m) * FPAD + lane) << 1) + (tt & 1)] = val;
    }
    __syncthreads();

#pragma unroll
    for (int ks = 0; ks < 4; ++ks) {
      const int kp = ks * 2 + kphalf;    // this half-wave's t-pair index
#pragma unroll
      for (int ff = 0; ff < 4; ++ff) {
        const int fl = wave + ff * 8;    // f_local within tile (zeros if f OOB)
        // single b64 load: (A[m][2*kp], A[m][2*kp+1])
        const v2f a = *(const v2f*)&tile[((kp * 16 + mlane) * FPAD + fl) << 1];
        // Gram: A-frag == B-frag layout, D = A*A^T + C
        acc[ff] = __builtin_amdgcn_wmma_f32_16x16x4_f32(
            false, a, false, a, (short)0, acc[ff], false, false);
      }
    }
    __syncthreads();
  }

  // store transposed (G symmetric): lane's 8 values are contiguous -> b128 x2
  const int col    = lane & 15;
  const int rowoff = (lane >> 4) << 3;
#pragma unroll
  for (int ff = 0; ff < 4; ++ff) {
    const int f = f0 + wave + ff * 8;
    if (f < FF) {
      float* dst = G + ((size_t)(n * FF) + f) * 256 + col * 16 + rowoff;
      const v4f* src = (const v4f*)&acc[ff];
      *(v4f*)(dst)     = src[0];
      *(v4f*)(dst + 4) = src[1];
    }
  }
}

// ---------------------------------------------------------------------------
// Kernel 2: per (n,f): Phi from Gram blocks, diagonal loading, in-register
// complex Gauss-Jordan inverse, MVDR weights; stores conj(w) interleaved.
// ---------------------------------------------------------------------------
__global__ __launch_bounds__(64) void mvdr_weights(const float* __restrict__ G,
                                                   const float* __restrict__ sv,
                                                   float* __restrict__ cw) {
  const int idx = blockIdx.x * blockDim.x + threadIdx.x;
  if (idx >= NN * FF) return;
  const int n = idx / FF;
  const int f = idx % FF;
  const float invT = 1.0f / (float)TT;

  float ar[8][8], ai[8][8];
  const size_t gb = (size_t)idx * 256;
#pragma unroll
  for (int c = 0; c < 8; ++c)
#pragma unroll
    for (int d = 0; d < 8; ++d) {
      const float rr = G[gb + c * 16 + d];             // Vr_c . Vr_d
      const float ii = G[gb + (8 + c) * 16 + (8 + d)]; // Vi_c . Vi_d
      const float ir = G[gb + (8 + c) * 16 + d];       // Vi_c . Vr_d
      const float ri = G[gb + c * 16 + (8 + d)];       // Vr_c . Vi_d
      ar[c][d] = (rr + ii) * invT;
      ai[c][d] = (ir - ri) * invT;
    }
#pragma unroll
  for (int c = 0; c < 8; ++c) { ar[c][c] += MVDR_LOAD; ai[c][c] += MVDR_LOAD; }

  // in-place complex Gauss-Jordan inversion (no pivoting; Phi ~ I-dominant)
#pragma unroll
  for (int k = 0; k < 8; ++k) {
    const float pr = ar[k][k], pi = ai[k][k];
    const float s  = 1.0f / (pr * pr + pi * pi);
    const float ipr = pr * s, ipi = -pi * s;
#pragma unroll
    for (int j = 0; j < 8; ++j)
      if (j != k) {
        const float xr = ar[k][j], xi = ai[k][j];
        ar[k][j] = xr * ipr - xi * ipi;
        ai[k][j] = xr * ipi + xi * ipr;
      }
#pragma unroll
    for (int i = 0; i < 8; ++i)
      if (i != k) {
        const float fr = ar[i][k], fi = ai[i][k];
#pragma unroll
        for (int j = 0; j < 8; ++j)
          if (j != k) {
            ar[i][j] -= fr * ar[k][j] - fi * ai[k][j];
            ai[i][j] -= fr * ai[k][j] + fi * ar[k][j];
          }
        ar[i][k] = -(fr * ipr - fi * ipi);
        ai[i][k] = -(fr * ipi + fi * ipr);
      }
    ar[k][k] = ipr;
    ai[k][k] = ipi;
  }

  // steering vector d: sv[((n*2+ri)*F + f)*C + c]
  float dr[8], di[8];
#pragma unroll
  for (int c = 0; c < 8; ++c) {
    dr[c] = sv[((size_t)(n * 2 + 0) * FF + f) * 8 + c];
    di[c] = sv[((size_t)(n * 2 + 1) * FF + f) * 8 + c];
  }

  // w_nume = Phi^-1 d
  float wr[8], wi[8];
#pragma unroll
  for (int c = 0; c < 8; ++c) {
    float sr = 0.0f, si = 0.0f;
#pragma unroll
    for (int d = 0; d < 8; ++d) {
      sr += ar[c][d] * dr[d] - ai[c][d] * di[d];
      si += ar[c][d] * di[d] + ai[c][d] * dr[d];
    }
    wr[c] = sr;
    wi[c] = si;
  }
  // deno = d^H w_nume
  float er = 0.0f, ei = 0.0f;
#pragma unroll
  for (int c = 0; c < 8; ++c) {
    er += dr[c] * wr[c] + di[c] * wi[c];
    ei += dr[c] * wi[c] - di[c] * wr[c];
  }
  const float s2 = 1.0f / (er * er + ei * ei);
#pragma unroll
  for (int c = 0; c < 8; ++c) {
    const float fr = (wr[c] * er + wi[c] * ei) * s2;   // w = w_nume/deno
    const float fi = (wi[c] * er - wr[c] * ei) * s2;
    cw[(size_t)idx * 16 + 2 * c]     = fr;             // conj(w)
    cw[(size_t)idx * 16 + 2 * c + 1] = -fi;
  }
}

// ---------------------------------------------------------------------------
// Kernel 3: beamform  X[n,t,f] = sum_c conj(w)[n,f,c] * y[n,c,t,f]
// ---------------------------------------------------------------------------
__global__ __launch_bounds__(256) void mvdr_apply(const float* __restrict__ mix,
                                                  const float* __restrict__ cw,
                                                  float* __restrict__ out) {
  const int idx = blockIdx.x * blockDim.x + threadIdx.x;
  if (idx >= NN * TT * FF) return;
  const int f = idx % FF;
  const int t = (idx / FF) % TT;
  const int n = idx / (FF * TT);
  float xr = 0.0f, xi = 0.0f;
  const size_t cb = ((size_t)(n * FF) + f) * 16;
#pragma unroll
  for (int c = 0; c < 8; ++c) {
    const float yr = mix[((size_t)(n * 16 + c) * TT + t) * FF + f];
    const float yi = mix[((size_t)(n * 16 + 8 + c) * TT + t) * FF + f];
    const float cr = cw[cb + 2 * c];
    const float ci = cw[cb + 2 * c + 1];
    xr += cr * yr - ci * yi;
    xi += cr * yi + ci * yr;
  }
  out[((size_t)(n * 2 + 0) * TT + t) * FF + f] = xr;
  out[((size_t)(n * 2 + 1) * TT + t) * FF + f] = xi;
}

extern "C" void kernel_launch(void* const* d_in, const int* in_sizes, int n_in,
                              void* d_out, int out_size, void* d_ws, size_t ws_size,
                              hipStream_t stream) {
  const float* mixture = (const float*)d_in[0];   // [8,2,8,600,257]
  const float* noise   = (const float*)d_in[1];   // [8,2,8,600,257]
  const float* sv      = (const float*)d_in[2];   // [8,2,257,8,1]
  float* out = (float*)d_out;                     // [8,2,1,600,257]
  float* ws  = (float*)d_ws;

  float* G  = ws;                         // 2056 * 256 floats  (~2.1 MB)
  float* cw = ws + (size_t)NN * FF * 256; // 2056 * 16 floats

  mvdr_cov<<<dim3(NN * NFT), dim3(256), 0, stream>>>(noise, G);
  mvdr_weights<<<dim3((NN * FF + 63) / 64), dim3(64), 0, stream>>>(G, sv, cw);
  mvdr_apply<<<dim3((NN * TT * FF + 255) / 256), dim3(256), 0, stream>>>(mixture, cw, out);
}